// LaplacianBuilder_52991306498405
// MI455X (gfx1250) — compile-verified
//
#include <hip/hip_runtime.h>
#include <hip/hip_bf16.h>

typedef __attribute__((ext_vector_type(2))) float v2f;
typedef __attribute__((ext_vector_type(4))) float v4f;
typedef __attribute__((ext_vector_type(8))) float v8f;

#define NNODES 1024
#define NEDGES 16384
#define LD     (NNODES * 16)   // 16384 row stride of output

// ---------------------------------------------------------------------------
// Pass 1: zero the 1 GiB output with 128-bit non-temporal stores.
// 1 GiB >> 192 MB L2, so NT is the correct temporal hint; this pass is the
// roofline bound (~46 us at 23.3 TB/s).
// ---------------------------------------------------------------------------
__global__ void __launch_bounds__(256)
zero_fill_kernel(v4f* __restrict__ out, size_t n4) {
    size_t i      = (size_t)blockIdx.x * blockDim.x + threadIdx.x;
    size_t stride = (size_t)gridDim.x * blockDim.x;
    v4f z = {};
    for (; i < n4; i += stride)
        __builtin_nontemporal_store(z, out + i);
}

// ---------------------------------------------------------------------------
// Pass 2: one wave32 per directed edge e in [0, 2E).
//   - diag contribution:  M_e^T * M_e  (4x V_WMMA_F32_16X16X4_F32, K=16)
//     scaled by dinv[src]^2, atomically added into block (src,src).
//   - if e < E: triu block M_from^T * M_to, scaled by -dinv[row]*dinv[col],
//     stored to block (row,col) and its transpose to (col,row).
//
// WMMA f32 16x16x4 operand layouts (ISA 7.12.2):
//   A (16x4, MxK): lane m=lane&15 holds M=m; VGPR v holds K = v + 2*(lane>>4)
//   B (4x16, KxN): lane n=lane&15 holds N=n; VGPR v holds K = v + 2*(lane>>4)
//   C/D (16x16):   VGPR r: M = r + 8*(lane>>4), N = lane&15
// A_w[m,k] = maps[e][k][m]  (transposed read), B_w[k,n] = maps[eB][k][n].
// ---------------------------------------------------------------------------
__global__ void __launch_bounds__(256)
edge_blocks_kernel(const float* __restrict__ maps,
                   const long long* __restrict__ edge_index,
                   const float* __restrict__ degrees,
                   float* __restrict__ out) {
    const int wave = threadIdx.x >> 5;
    const int lane = threadIdx.x & 31;
    const int e    = blockIdx.x * 8 + wave;   // directed edge id, [0, 2E)
    const int m    = lane & 15;
    const int hi   = lane >> 4;

    const float* __restrict__ mapsE = maps + (size_t)e * 256;

    // A = maps[e]^T in the 16x4 f32 A layout, 4 K-chunks of 4
    v2f a[4], b[4];
#pragma unroll
    for (int kk = 0; kk < 4; ++kk) {
        const int k0 = kk * 4 + 2 * hi;
        a[kk].x = mapsE[(k0 + 0) * 16 + m];
        a[kk].y = mapsE[(k0 + 1) * 16 + m];
        // B = maps[e] in the 4x16 B layout (same index math, n == m)
        b[kk].x = a[kk].x;
        b[kk].y = a[kk].y;
    }

    v8f cdiag = {};
#pragma unroll
    for (int kk = 0; kk < 4; ++kk)
        cdiag = __builtin_amdgcn_wmma_f32_16x16x4_f32(
            false, a[kk], false, b[kk], (short)0, cdiag, false, false);

    const int   src      = (int)edge_index[e];                 // edge_index[0][e]
    const float dinv_src = rsqrtf(degrees[src] * 16.0f + 1.0f);
    const float s2       = dinv_src * dinv_src;

#pragma unroll
    for (int r = 0; r < 8; ++r) {
        const int M = r + 8 * hi;
        const size_t addr = (size_t)(src * 16 + M) * LD + (size_t)(src * 16 + m);
        unsafeAtomicAdd(out + addr, s2 * cdiag[r]);            // global_atomic_add_f32
    }

    if (e < NEDGES) {   // wave-uniform branch: EXEC stays all-ones for WMMA
        const float* __restrict__ mapsB = maps + ((size_t)e + NEDGES) * 256;
        v2f bb[4];
#pragma unroll
        for (int kk = 0; kk < 4; ++kk) {
            const int k0 = kk * 4 + 2 * hi;
            bb[kk].x = mapsB[(k0 + 0) * 16 + m];
            bb[kk].y = mapsB[(k0 + 1) * 16 + m];
        }
        v8f ct = {};
#pragma unroll
        for (int kk = 0; kk < 4; ++kk)
            ct = __builtin_amdgcn_wmma_f32_16x16x4_f32(
                false, a[kk], false, bb[kk], (short)0, ct, false, false);

        const int   row = src;
        const int   col = (int)edge_index[2 * NEDGES + e];     // edge_index[1][e]
        const float s   = -(dinv_src * rsqrtf(degrees[col] * 16.0f + 1.0f));

#pragma unroll
        for (int r = 0; r < 8; ++r) {
            const int   M = r + 8 * hi;
            const float v = s * ct[r];
            out[(size_t)(row * 16 + M) * LD + (size_t)(col * 16 + m)] = v;   // -T
            out[(size_t)(col * 16 + m) * LD + (size_t)(row * 16 + M)] = v;   // -T^T
        }
    }
}

extern "C" void kernel_launch(void* const* d_in, const int* in_sizes, int n_in,
                              void* d_out, int out_size, void* d_ws, size_t ws_size,
                              hipStream_t stream) {
    // setup_inputs order: adj_mat (unused), degrees, maps, edge_index (int64)
    const float*     degrees    = (const float*)d_in[1];
    const float*     maps       = (const float*)d_in[2];
    const long long* edge_index = (const long long*)d_in[3];
    float*           out        = (float*)d_out;

    const size_t n4 = (size_t)out_size / 4;                    // 2^26 float4s
    zero_fill_kernel<<<8192, 256, 0, stream>>>((v4f*)out, n4);

    // 2E = 32768 waves, 8 waves (256 threads) per block -> 4096 blocks
    edge_blocks_kernel<<<(2 * NEDGES) / 8, 256, 0, stream>>>(maps, edge_index,
                                                             degrees, out);
}